// Model_20624432955428
// MI455X (gfx1250) — compile-verified
//
#include <hip/hip_runtime.h>

typedef __attribute__((ext_vector_type(2))) float v2f;
typedef __attribute__((ext_vector_type(8))) float v8f;
typedef __attribute__((ext_vector_type(4))) unsigned int v4u;
typedef __attribute__((ext_vector_type(8))) int v8i;
typedef __attribute__((ext_vector_type(4))) int v4i;

#define T_TOTAL 262144
#define GRP     8
#define KDIM    512
#define NDIM    512
#define TG      (T_TOTAL / GRP)   // 32768 rows per group
#define TILE_M  128
#define TILE_N  128
#define KB      32                // X K-block per TDM tile
#define XS_STRIDE 36              // 32 dwords + 4 pad (TDM pad_interval=32dw, pad_amount=4dw)
#define WT_STRIDE 546             // transposed W: [128 cols][512 k + 34 pad]; pair-stride 273 ≡ 17 (mod 32)

// ---- Tensor Data Mover: 2D tile load X[128 rows x 32 cols] -> LDS (padded rows) ----
// D# per CDNA5 ISA §8.3/8.4: group0 = {count/type/addr}, group1 = dims/strides/pad.
__device__ __forceinline__ void tdm_load_x_tile(const float* gptr, unsigned lds_off) {
  unsigned long long ga = (unsigned long long)(uintptr_t)gptr;
  v4u g0;
  g0.x = 1u;                                         // count=1, user mode, no gather
  g0.y = lds_off;                                    // lds_addr (bytes)
  g0.z = (unsigned)(ga & 0xFFFFFFFFu);               // global_addr[31:0]
  g0.w = (unsigned)((ga >> 32) & 0x1FFFFFFu)         // global_addr[56:32]
       | (2u << 30);                                 // type=2 ("image")
  v8i g1;
  g1[0] = (2 << 16)                                  // data_size = 4B
        | (1 << 20)                                  // pad_enable
        | (4 << 22)                                  // pad_interval: 32 dwords
        | (3 << 25);                                 // pad_amount: 4 dwords -> LDS row stride 36
  g1[1] = (int)(512u << 16);                         // tensor_dim0[15:0]=512 (bits 79:48 span)
  g1[2] = 0;                                         // tensor_dim0[31:16]=0, tensor_dim1[15:0]=0
  g1[3] = 4 | (KB << 16);                            // tensor_dim1[31:16]=4 (262144), tile_dim0=32
  g1[4] = TILE_M;                                    // tile_dim1=128, tile_dim2=0
  g1[5] = 512;                                       // tensor_dim0_stride lo
  g1[6] = 0;                                         // stride hi / dim1_stride lo (unused, 2D)
  g1[7] = 0;
  v4i gz4 = {};
  v8i gz8 = {};
  __builtin_amdgcn_tensor_load_to_lds(g0, g1, gz4, gz4, gz8, 0);
}

__global__ __launch_bounds__(256) void grouped_gemm_wmma_tdm(
    const float* __restrict__ x,     // [T_TOTAL, K]
    const float* __restrict__ w,     // [G, K, N]
    const float* __restrict__ bias,  // [G, N]
    float* __restrict__ out)         // [T_TOTAL, N]
{
  __shared__ float Xs[2][TILE_M * XS_STRIDE];   // 2 x 18 KB, filled by TDM
  __shared__ float Wt[TILE_N * WT_STRIDE];      // 273 KB: whole transposed W slab [n][k]

  const int tid    = threadIdx.x;
  const int wave   = tid >> 5;
  const int lane   = tid & 31;
  const int h      = lane >> 4;   // lane-half: K sub-pair select / rows +8
  const int ln     = lane & 15;
  const int wave_m = wave >> 1;   // 0..3 -> 32-row slab
  const int wave_n = wave & 1;    // 0..1 -> 64-col slab

  const int blk      = blockIdx.x;
  const int mtile    = blk >> 2;                 // N-tile fastest: X tile L2 reuse
  const int ntile    = blk & 3;
  const int g        = mtile / (TG / TILE_M);
  const int row_base = mtile * TILE_M;
  const int col_base = ntile * TILE_N;

  const float* __restrict__ wg = w + (size_t)g * KDIM * NDIM;
  const float* __restrict__ bg = bias + g * NDIM;
  const float* __restrict__ xg = x + (size_t)row_base * KDIM;

  // ---- prologue: kick off first X tile DMA, then stage W under it ----
  if (wave == 0)
    tdm_load_x_tile(xg, (unsigned)(uintptr_t)&Xs[0][0]);

  // Stage entire W column-slab transposed: Wt[n][k] = W[k][col_base+n].
  // Read coalesced k-row pairs, write b64 (k,k+1) pairs -> B frags become single b64 loads.
  {
    const int kp = tid >> 4;           // k-pair index within a 32-row chunk
    const int nc = (tid & 15) * 8;     // 8-column chunk
    for (int kb = 0; kb < KDIM; kb += KB) {
      const int kk = kb + 2 * kp;
      const float* s0 = wg + (size_t)kk * NDIM + col_base + nc;
      const float* s1 = s0 + NDIM;
      float4 q0 = *(const float4*)(s0);
      float4 q1 = *(const float4*)(s0 + 4);
      float4 q2 = *(const float4*)(s1);
      float4 q3 = *(const float4*)(s1 + 4);
      float r0[8] = {q0.x, q0.y, q0.z, q0.w, q1.x, q1.y, q1.z, q1.w};
      float r1[8] = {q2.x, q2.y, q2.z, q2.w, q3.x, q3.y, q3.z, q3.w};
#pragma unroll
      for (int i = 0; i < 8; ++i) {
        v2f p;
        p.x = r0[i];
        p.y = r1[i];
        *(v2f*)&Wt[(nc + i) * WT_STRIDE + kk] = p;  // 8B-aligned (546 even, kk even)
      }
    }
  }
  __syncthreads();  // W slab visible to all waves

  v8f zero = {};
  v8f acc[2][4];
#pragma unroll
  for (int mt = 0; mt < 2; ++mt)
#pragma unroll
    for (int nt = 0; nt < 4; ++nt)
      acc[mt][nt] = zero;

  // ---- main loop: double-buffered TDM X stream + WMMA ----
  int cur = 0;
  for (int kb = 0; kb < KDIM; kb += KB) {
    if (wave == 0)
      __builtin_amdgcn_s_wait_tensorcnt(0);  // X[cur] DMA complete
    __syncthreads();                          // publish X[cur]; all waves done with X[cur^1]
    if (wave == 0 && kb + KB < KDIM)
      tdm_load_x_tile(xg + (kb + KB),
                      (unsigned)(uintptr_t)&Xs[cur ^ 1][0]);  // overlaps compute below

    const float* xsb = &Xs[cur][0];
#pragma unroll
    for (int k0 = 0; k0 < KB; k0 += 4) {
      // A frag (16x4 f32): lane 16h+m holds K = k0+2h, k0+2h+1  -> one b64
      v2f a[2];
#pragma unroll
      for (int mt = 0; mt < 2; ++mt) {
        const int row = wave_m * 32 + mt * 16 + ln;
        a[mt] = *(const v2f*)&xsb[row * XS_STRIDE + k0 + 2 * h];
      }
      // B frag (4x16 f32): lane 16h+n holds W[kb+k0+2h][col], W[kb+k0+2h+1][col] -> one b64
      v2f b[4];
#pragma unroll
      for (int nt = 0; nt < 4; ++nt) {
        const int col = wave_n * 64 + nt * 16 + ln;
        b[nt] = *(const v2f*)&Wt[col * WT_STRIDE + kb + k0 + 2 * h];
      }
#pragma unroll
      for (int mt = 0; mt < 2; ++mt)
#pragma unroll
        for (int nt = 0; nt < 4; ++nt)
          acc[mt][nt] = __builtin_amdgcn_wmma_f32_16x16x4_f32(
              false, a[mt], false, b[nt], (short)0, acc[mt][nt], false, false);
    }
    cur ^= 1;
  }

  // ---- epilogue: bias add + store (C/D layout: VGPR r -> rows r / r+8 per lane half) ----
#pragma unroll
  for (int mt = 0; mt < 2; ++mt) {
    const int mrow = row_base + wave_m * 32 + mt * 16 + 8 * h;
#pragma unroll
    for (int nt = 0; nt < 4; ++nt) {
      const int col = col_base + wave_n * 64 + nt * 16 + ln;
      const float bv = bg[col];
      v8f c = acc[mt][nt];
#pragma unroll
      for (int r = 0; r < 8; ++r)
        out[(size_t)(mrow + r) * NDIM + col] = c[r] + bv;
    }
  }
}

extern "C" void kernel_launch(void* const* d_in, const int* in_sizes, int n_in,
                              void* d_out, int out_size, void* d_ws, size_t ws_size,
                              hipStream_t stream) {
  const float* x    = (const float*)d_in[0];  // [262144, 512]
  const float* w    = (const float*)d_in[1];  // [8, 512, 512]
  const float* bias = (const float*)d_in[2];  // [8, 512]
  float*       out  = (float*)d_out;          // [262144, 512]

  const int num_blocks = (T_TOTAL / TILE_M) * (NDIM / TILE_N);  // 8192
  grouped_gemm_wmma_tdm<<<dim3(num_blocks), dim3(256), 0, stream>>>(x, w, bias, out);
}